// SpatialAttentionBlock3d_14568529068391
// MI455X (gfx1250) — compile-verified
//
#include <hip/hip_runtime.h>
#include <hip/hip_bf16.h>

typedef float v2f __attribute__((ext_vector_type(2)));
typedef float v8f __attribute__((ext_vector_type(8)));

#define NN 4096          // 16*16*16 spatial points
#define CB 64            // channels
#define CQ 8             // reduced channels
#define BB 2             // batch

// ---------------------------------------------------------------------------
// K1: fused tiny convs. One thread per (b, n). Writes qT/kT/jT as [b][n][8]
// (transposed, so later 8-dots load contiguously) and v as [b][c][n].
// ---------------------------------------------------------------------------
__global__ void k_conv(const float* __restrict__ x,
                       const float* __restrict__ wq, const float* __restrict__ bq,
                       const float* __restrict__ wk, const float* __restrict__ bk,
                       const float* __restrict__ wj, const float* __restrict__ bj,
                       const float* __restrict__ wv, const float* __restrict__ bv,
                       float* __restrict__ qT, float* __restrict__ kT,
                       float* __restrict__ jT, float* __restrict__ v)
{
    int t = blockIdx.x * blockDim.x + threadIdx.x;   // b*NN + n
    if (t >= BB * NN) return;
    int b  = t >> 12;
    int n  = t & (NN - 1);
    int s2 = n & 15, s1 = (n >> 4) & 15, s0 = n >> 8;

    const float* xb = x + (size_t)b * CB * NN;

    float qa[CQ], ka[CQ], ja[CQ];
#pragma unroll
    for (int i = 0; i < CQ; ++i) { qa[i] = bq[i]; ka[i] = bk[i]; ja[i] = bj[i]; }
    float va[CB];
#pragma unroll
    for (int c = 0; c < CB; ++c) va[c] = bv[c];

    for (int ci = 0; ci < CB; ++ci) {
        const float* xc = xb + (size_t)ci * NN + n;
        float x0  = xc[0];
        float xqm = (s1 > 0)  ? xc[-16]  : 0.f;   // 3-tap along s1 (wq)
        float xqp = (s1 < 15) ? xc[ 16]  : 0.f;
        float xkm = (s0 > 0)  ? xc[-256] : 0.f;   // 3-tap along s0 (wk)
        float xkp = (s0 < 15) ? xc[ 256] : 0.f;
        float xjm = (s2 > 0)  ? xc[-1]   : 0.f;   // 3-tap along s2 (wj)
        float xjp = (s2 < 15) ? xc[ 1]   : 0.f;
#pragma unroll
        for (int cq = 0; cq < CQ; ++cq) {
            const float* w = wq + (cq * CB + ci) * 3;
            qa[cq] += w[0] * xqm + w[1] * x0 + w[2] * xqp;
            w = wk + (cq * CB + ci) * 3;
            ka[cq] += w[0] * xkm + w[1] * x0 + w[2] * xkp;
            w = wj + (cq * CB + ci) * 3;
            ja[cq] += w[0] * xjm + w[1] * x0 + w[2] * xjp;
        }
#pragma unroll
        for (int c = 0; c < CB; ++c) va[c] += wv[c * CB + ci] * x0;
    }

    float* qo = qT + (size_t)t * CQ;
    float* ko = kT + (size_t)t * CQ;
    float* jo = jT + (size_t)t * CQ;
#pragma unroll
    for (int i = 0; i < CQ; ++i) { qo[i] = qa[i]; ko[i] = ka[i]; jo[i] = ja[i]; }
#pragma unroll
    for (int c = 0; c < CB; ++c) v[((size_t)(b * CB + c)) * NN + n] = va[c];
}

// ---------------------------------------------------------------------------
// K2: kj[b][c][d] = sum_n k[c,n]*j[d,n].   256 threads, LDS reduction.
// ---------------------------------------------------------------------------
__global__ void k_kj(const float* __restrict__ kT, const float* __restrict__ jT,
                     float* __restrict__ kj)
{
    __shared__ float part[4][64];
    int b   = blockIdx.x;
    int tid = threadIdx.x;
    int cd  = tid & 63;
    int c   = cd >> 3, d = cd & 7;
    int s   = tid >> 6;                       // n-slice 0..3
    const float* kb = kT + (size_t)b * NN * CQ;
    const float* jb = jT + (size_t)b * NN * CQ;
    float acc = 0.f;
    for (int n = s * 1024; n < (s + 1) * 1024; ++n)
        acc += kb[n * CQ + c] * jb[n * CQ + d];
    part[s][cd] = acc;
    __syncthreads();
    if (tid < 64)
        kj[b * 64 + tid] = part[0][tid] + part[1][tid] + part[2][tid] + part[3][tid];
}

// ---------------------------------------------------------------------------
// K3: pT[b][n][d] = sum_c kj[b][c][d] * q[c,n].  One thread per (b,n).
// Then aff[n,m] = dot8(pT[n], kT[m]).
// ---------------------------------------------------------------------------
__global__ void k_p(const float* __restrict__ qT, const float* __restrict__ kj,
                    float* __restrict__ pT)
{
    int t = blockIdx.x * blockDim.x + threadIdx.x;   // b*NN + n
    if (t >= BB * NN) return;
    int b = t >> 12;
    const float* q = qT + (size_t)t * CQ;
    float qq[CQ];
#pragma unroll
    for (int i = 0; i < CQ; ++i) qq[i] = q[i];
    const float* K = kj + b * 64;
    float* po = pT + (size_t)t * CQ;
#pragma unroll
    for (int d = 0; d < CQ; ++d) {
        float a = 0.f;
#pragma unroll
        for (int c = 0; c < CQ; ++c) a += K[c * CQ + d] * qq[c];
        po[d] = a;
    }
}

__device__ __forceinline__ float dot8(const float* a, const float* b) {
    float s = 0.f;
#pragma unroll
    for (int i = 0; i < 8; ++i) s += a[i] * b[i];
    return s;
}

// ---------------------------------------------------------------------------
// K4: per-row softmax stats. One wave32 per row n: rowmax[n], 1/rowsum[n].
// ---------------------------------------------------------------------------
__global__ void k_rowstats(const float* __restrict__ pT, const float* __restrict__ kT,
                           float* __restrict__ rmax, float* __restrict__ rrcp)
{
    int wv   = (blockIdx.x * blockDim.x + threadIdx.x) >> 5;  // b*NN + n
    int lane = threadIdx.x & 31;
    if (wv >= BB * NN) return;
    int b = wv >> 12;

    float p[8];
    const float* pn = pT + (size_t)wv * CQ;
#pragma unroll
    for (int i = 0; i < 8; ++i) p[i] = pn[i];
    const float* kb = kT + (size_t)b * NN * CQ;

    float mx = -3.4e38f;
    for (int m = lane; m < NN; m += 32)
        mx = fmaxf(mx, dot8(p, kb + m * CQ));
#pragma unroll
    for (int off = 16; off >= 1; off >>= 1)
        mx = fmaxf(mx, __shfl_xor(mx, off, 32));

    float s = 0.f;
    for (int m = lane; m < NN; m += 32)
        s += __expf(dot8(p, kb + m * CQ) - mx);
#pragma unroll
    for (int off = 16; off >= 1; off >>= 1)
        s += __shfl_xor(s, off, 32);

    if (lane == 0) { rmax[wv] = mx; rrcp[wv] = 1.f / s; }
}

// ---------------------------------------------------------------------------
// K5: vhat[c,n] = v[c,n] / rowsum[n]  (fold softmax denominator into V).
// ---------------------------------------------------------------------------
__global__ void k_vhat(const float* __restrict__ v, const float* __restrict__ rrcp,
                       float* __restrict__ vh)
{
    int i = blockIdx.x * blockDim.x + threadIdx.x;   // over B*CB*NN
    if (i >= BB * CB * NN) return;
    int b = i >> 18;                                  // / (CB*NN)
    int n = i & (NN - 1);
    vh[i] = v[i] * rrcp[b * NN + n];
}

// ---------------------------------------------------------------------------
// K6: fused attention-apply GEMM:
//   out[c,m] = gamma * sum_n vhat[c,n] * exp(dot8(p_n,k_m) - rowmax[n]) + x[c,m]
// Block = 128 threads (4 waves), one 32-column m-tile per block.
// Per 32-row n-chunk: 128 threads build E[32][32] in LDS, then each wave
// (owning C rows [16w,16w+16)) runs 16x wmma_f32_16x16x4_f32 with TWO
// accumulators (m-halves), reusing each A fragment twice. A/B fragments are
// preloaded per chunk so global loads clause up and WMMAs issue back-to-back
// as two independent dependency chains.
// ---------------------------------------------------------------------------
__global__ void k_attn(const float* __restrict__ vh, const float* __restrict__ pT,
                       const float* __restrict__ kT, const float* __restrict__ rmax,
                       const float* __restrict__ x, const float* __restrict__ gamma,
                       float* __restrict__ out)
{
    __shared__ float E[32][33];                  // padded: bank-conflict free

    int b    = blockIdx.x >> 7;                  // NN/32 = 128 tiles per batch
    int m0   = (blockIdx.x & 127) << 5;
    int tid  = threadIdx.x;
    int lane = tid & 31;
    int wv   = tid >> 5;                         // wave id 0..3 -> C block
    int col  = tid & 31;                         // E-producer column (0..31)
    int grp  = tid >> 5;                         // E-producer row group 0..3
    int row  = lane & 15;                        // WMMA fragment row/col
    int hi   = lane >> 4;                        // WMMA half-wave (K split)

    const float* kB  = kT   + (size_t)b * NN * CQ;
    const float* pB  = pT   + (size_t)b * NN * CQ;
    const float* rmB = rmax + (size_t)b * NN;

    // this thread's k-column (fixed for whole n loop)
    float kc[8];
    {
        const float* km = kB + (size_t)(m0 + col) * CQ;
#pragma unroll
        for (int i = 0; i < 8; ++i) kc[i] = km[i];
    }

    v8f acc0 = {}, acc1 = {};
    // A-fragment base: vhat row (c0+row), K offset hi*2 per ISA 16x4 layout
    const float* arow = vh + ((size_t)(b * CB + wv * 16 + row)) * NN + hi * 2;

    for (int n0 = 0; n0 < NN; n0 += 32) {
        // ---- cooperative E[32][32] = exp(aff - rowmax), 8 elems/thread ----
#pragma unroll
        for (int r = 0; r < 8; ++r) {
            int nl = grp * 8 + r;
            int n  = n0 + nl;
            float a = dot8(pB + (size_t)n * CQ, kc);
            E[nl][col] = __expf(a - rmB[n]);
        }
        __syncthreads();

        // ---- preload all fragments for this chunk ----
        v2f a[8], b0[8], b1[8];
#pragma unroll
        for (int kk = 0; kk < 8; ++kk)
            a[kk] = *(const v2f*)(arow + n0 + kk * 4);           // A: 16x4 f32
#pragma unroll
        for (int kk = 0; kk < 8; ++kk) {
            int kb2 = kk * 4 + hi * 2;
            b0[kk].x = E[kb2][row];          b0[kk].y = E[kb2 + 1][row];
            b1[kk].x = E[kb2][row + 16];     b1[kk].y = E[kb2 + 1][row + 16];
        }

        // ---- 16 x WMMA f32 16x16x4, two independent accumulator chains ----
#pragma unroll
        for (int kk = 0; kk < 8; ++kk) {
            acc0 = __builtin_amdgcn_wmma_f32_16x16x4_f32(
                false, a[kk], false, b0[kk], (short)0, acc0, false, false);
            acc1 = __builtin_amdgcn_wmma_f32_16x16x4_f32(
                false, a[kk], false, b1[kk], (short)0, acc1, false, false);
        }
        __syncthreads();
    }

    // ---- epilogue: out = gamma*acc + x (residual) ----
    float g = gamma[0];
#pragma unroll
    for (int r = 0; r < 8; ++r) {
        int c = wv * 16 + r + hi * 8;                    // C/D VGPR layout
        size_t idx = ((size_t)(b * CB + c)) * NN + m0 + row;
        out[idx]      = g * acc0[r] + x[idx];
        out[idx + 16] = g * acc1[r] + x[idx + 16];
    }
}

// ---------------------------------------------------------------------------
extern "C" void kernel_launch(void* const* d_in, const int* in_sizes, int n_in,
                              void* d_out, int out_size, void* d_ws, size_t ws_size,
                              hipStream_t stream) {
    const float* x     = (const float*)d_in[0];
    const float* wq    = (const float*)d_in[1];
    const float* bq    = (const float*)d_in[2];
    const float* wk    = (const float*)d_in[3];
    const float* bk    = (const float*)d_in[4];
    const float* wj    = (const float*)d_in[5];
    const float* bj    = (const float*)d_in[6];
    const float* wv    = (const float*)d_in[7];
    const float* bv    = (const float*)d_in[8];
    const float* gamma = (const float*)d_in[9];
    float* out = (float*)d_out;

    float* ws  = (float*)d_ws;                 // ~5.1 MB total
    float* qT  = ws;                           // BB*NN*8   = 65536
    float* kT  = ws + 65536;                   // 65536
    float* jT  = ws + 131072;                  // 65536
    float* pT  = ws + 196608;                  // 65536
    float* v   = ws + 262144;                  // BB*CB*NN  = 524288
    float* vh  = ws + 786432;                  // 524288
    float* kj  = ws + 1310720;                 // 128
    float* rmx = ws + 1310848;                 // 8192
    float* rrc = ws + 1319040;                 // 8192

    k_conv    <<<dim3((BB * NN + 255) / 256), dim3(256), 0, stream>>>(
                  x, wq, bq, wk, bk, wj, bj, wv, bv, qT, kT, jT, v);
    k_kj      <<<dim3(BB),   dim3(256), 0, stream>>>(kT, jT, kj);
    k_p       <<<dim3(32),   dim3(256), 0, stream>>>(qT, kj, pT);
    k_rowstats<<<dim3(1024), dim3(256), 0, stream>>>(pT, kT, rmx, rrc);
    k_vhat    <<<dim3((BB * CB * NN + 255) / 256), dim3(256), 0, stream>>>(v, rrc, vh);
    k_attn    <<<dim3(BB * (NN / 32)), dim3(128), 0, stream>>>(
                  vh, pT, kT, rmx, x, gamma, out);
}